// KPConvD_16157666968108
// MI455X (gfx1250) — compile-verified
//
#include <hip/hip_runtime.h>
#include <math.h>

#define HN   32      // neighbors per query
#define CC   64      // channels
#define KP   15      // kernel points
#define QPB  8       // queries per block (one wave32 per query)
#define SIGMA 0.3f

typedef unsigned int u32x4 __attribute__((ext_vector_type(4)));
typedef int          i32x8 __attribute__((ext_vector_type(8)));
typedef int          i32x4 __attribute__((ext_vector_type(4)));

__global__ __launch_bounds__(256)
void kpconv_depthwise_kernel(const float* __restrict__ q_pts,
                             const float* __restrict__ s_pts,
                             const float* __restrict__ s_feats,
                             const int*   __restrict__ neighb_inds,
                             const float* __restrict__ weights,
                             const float* __restrict__ kpts,
                             float* __restrict__ out,
                             int N)
{
    __shared__ float w_lds[KP * CC];      // 3840 B weight tile, filled by TDM

    const int tid = threadIdx.x;

    // Kernel points: uniform read-only addresses -> compiler scalarizes to s_load,
    // values live in SGPRs (no LDS, no per-lane vector loads).
    float kx[KP], ky[KP], kz[KP];
#pragma unroll
    for (int k = 0; k < KP; ++k) {
        kx[k] = kpts[3 * k + 0];
        ky[k] = kpts[3 * k + 1];
        kz[k] = kpts[3 * k + 2];
    }

#if __has_builtin(__builtin_amdgcn_tensor_load_to_lds)
    // TDM: DMA weights[15][64] f32 -> LDS, issued once by wave 0.
    if ((tid >> 5) == 0) {
        unsigned long long ga = (unsigned long long)(const void*)weights;
        unsigned int lds_off  = (unsigned int)(unsigned long long)&w_lds[0];

        u32x4 g0;
        g0.x = 1u;                                   // count=1 valid descriptor, user mode
        g0.y = lds_off;                              // lds_addr [63:32]
        g0.z = (unsigned int)ga;                     // global_addr [95:64]
        g0.w = ((unsigned int)(ga >> 32) & 0x01FFFFFFu)
             | 0x80000000u;                          // global_addr[56:32] | type=2

        i32x8 g1;
        g1[0] = (int)(2u << 16);                     // workgroup_mask=0, data_size=2 (4B)
        g1[1] = (int)((64u & 0xFFFFu) << 16);        // tensor_dim0=64 (low16 @ bits63:48)
        g1[2] = (int)((64u >> 16) | ((15u & 0xFFFFu) << 16)); // dim0 hi | tensor_dim1=15 lo
        g1[3] = (int)((15u >> 16) | (64u << 16));    // tensor_dim1 hi | tile_dim0=64
        g1[4] = (int)(15u);                          // tile_dim1=15, tile_dim2=0
        g1[5] = 64;                                  // tensor_dim0_stride low32 = 64
        g1[6] = 0;
        g1[7] = 0;
        i32x4 z4 = {0, 0, 0, 0};
        i32x8 z8 = {0, 0, 0, 0, 0, 0, 0, 0};

        __builtin_amdgcn_tensor_load_to_lds(g0, g1, z4, z4, z8, 0);
#if __has_builtin(__builtin_amdgcn_s_wait_tensorcnt)
        __builtin_amdgcn_s_wait_tensorcnt(0);
#else
        asm volatile("s_wait_tensorcnt 0" ::: "memory");
#endif
    }
#else
    for (int i = tid; i < KP * CC; i += 256) w_lds[i] = weights[i];
#endif
    __syncthreads();   // publish w_lds to all 8 waves

    // ---------------- Phase 1: lane h owns neighbor h of this wave's query ----
    const int qi = tid >> 5;                 // wave id == query slot
    const int h  = tid & 31;
    const int m  = blockIdx.x * QPB + qi;

    int   packr = 0;      // idx | (bk << 20)
    float inflr = 0.0f;
    if (m < N) {
        int raw = neighb_inds[(size_t)m * HN + h];
        if (raw >= 0 && raw < N) {
            const float qx = q_pts[m * 3 + 0];
            const float qy = q_pts[m * 3 + 1];
            const float qz = q_pts[m * 3 + 2];
            const float dx = s_pts[(size_t)raw * 3 + 0] - qx;
            const float dy = s_pts[(size_t)raw * 3 + 1] - qy;
            const float dz = s_pts[(size_t)raw * 3 + 2] - qz;
            float best = 3.4e38f; int bk = 0;
#pragma unroll
            for (int k = 0; k < KP; ++k) {
                const float ax = dx - kx[k];
                const float ay = dy - ky[k];
                const float az = dz - kz[k];
                const float dd = ax * ax + ay * ay + az * az;
                if (dd < best) { best = dd; bk = k; }  // strict '<' => first argmin (jnp)
            }
            float t = 1.0f - sqrtf(best) * (1.0f / SIGMA);
            inflr = t > 0.0f ? t : 0.0f;
            packr = raw | (bk << 20);
            if (inflr > 0.0f) {
                // warm caches for the feature row phase 2 will gather
                __builtin_prefetch(&s_feats[(size_t)raw * CC], 0, 1);
                __builtin_prefetch(&s_feats[(size_t)raw * CC + 32], 0, 1);
            }
        }
    }

    // ---------------- Phase 2: same wave, broadcast via v_readlane ------------
    // 32 lanes x float2 = 64 channels. id/kr become SGPRs -> scalar-based gather
    // addressing and s_cbranch skip of dead neighbors. No LDS staging, no barrier.
    const int c = h * 2;
    float a0 = 0.0f, a1 = 0.0f;
#pragma unroll 8
    for (int hh = 0; hh < HN; ++hh) {
        const float wI = __int_as_float(
            __builtin_amdgcn_readlane(__float_as_int(inflr), hh));
        if (wI <= 0.0f) continue;                       // wave-uniform scalar skip
        const int pk = __builtin_amdgcn_readlane(packr, hh);
        const int id = pk & 0xFFFFF;
        const int kr = (pk >> 20) << 6;                 // bk * 64
        const float2 f = *(const float2*)(s_feats + (size_t)id * CC + c);
        const float2 w = *(const float2*)(w_lds + kr + c);
        a0 = fmaf(f.x, w.x * wI, a0);
        a1 = fmaf(f.y, w.y * wI, a1);
    }
    if (m < N) {
        *(float2*)(out + (size_t)m * CC + c) = make_float2(a0, a1);
    }
}

extern "C" void kernel_launch(void* const* d_in, const int* in_sizes, int n_in,
                              void* d_out, int out_size, void* d_ws, size_t ws_size,
                              hipStream_t stream) {
    const float* q_pts   = (const float*)d_in[0];
    const float* s_pts   = (const float*)d_in[1];
    const float* s_feats = (const float*)d_in[2];
    const int*   neighb  = (const int*)d_in[3];
    const float* weights = (const float*)d_in[4];
    const float* kpts    = (const float*)d_in[5];
    float* out = (float*)d_out;

    const int N = in_sizes[0] / 3;
    const int blocks = (N + QPB - 1) / QPB;
    hipLaunchKernelGGL(kpconv_depthwise_kernel, dim3(blocks), dim3(256), 0, stream,
                       q_pts, s_pts, s_feats, neighb, weights, kpts, out, N);
}